// LanguageMixer_19731079758016
// MI455X (gfx1250) — compile-verified
//
#include <hip/hip_runtime.h>
#include <hip/hip_bf16.h>
#include <stdint.h>

// out[:,2i]   = (a+b)   % 10 + 1
// out[:,2i+1] = (10+a-b)% 10 + 1   where a=inp[:,2i], b=inp[:,2i+1]
// Pure streaming: ~272 MB traffic -> ~12us at 23.3 TB/s. Use CDNA5 async
// global->LDS (ASYNCcnt) double-buffered pipeline + non-temporal stores.

typedef int   __attribute__((ext_vector_type(4))) i4;
typedef float __attribute__((ext_vector_type(4))) f4;

#define TPB        256
#define ITERS      8
#define TILE_INTS  (TPB * 4)        // 1024 ints = 4 KB per tile
#define TILE_BYTES (TILE_INTS * 4)

__global__ __launch_bounds__(TPB)
void mixer_main(const int* __restrict__ inp, float* __restrict__ out, int nvec) {
    // nvec = number of int4 groups (2 pairs each)
    __shared__ int lds[2][TILE_INTS];

    const int tid      = threadIdx.x;
    const int tile0    = blockIdx.x * ITERS;
    const uint32_t ldsBase = (uint32_t)(uintptr_t)(&lds[0][0]); // low 32b = LDS byte addr

    auto issue = [&](int it, int buf) {
        int v = (tile0 + it) * TPB + tid;          // int4 index
        if (v >= nvec) v = nvec - 1;               // clamp: harmless duplicate load
        uint32_t voff  = (uint32_t)v * 16u;        // byte offset into inp
        uint32_t laddr = ldsBase + (uint32_t)buf * TILE_BYTES + (uint32_t)tid * 16u;
        // CDNA5 async DMA: global -> LDS, tracked by ASYNCcnt (GVS mode)
        asm volatile("global_load_async_to_lds_b128 %0, %1, %2"
                     :: "v"(laddr), "v"(voff), "s"(inp)
                     : "memory");
    };

    int buf = 0;
    issue(0, 0);

#pragma unroll
    for (int it = 0; it < ITERS; ++it) {
        if (it + 1 < ITERS) {
            issue(it + 1, buf ^ 1);                       // prefetch next tile
            asm volatile("s_wait_asynccnt 0x1" ::: "memory"); // current tile done
        } else {
            asm volatile("s_wait_asynccnt 0x0" ::: "memory");
        }

        // Each lane reads back exactly the 16 bytes it requested: no barrier needed.
        i4 p = *(const i4*)((const char*)(&lds[0][0]) + buf * TILE_BYTES + tid * 16);

        int a0 = p.x, b0 = p.y, a1 = p.z, b1 = p.w;      // values in [0,9]
        int s0 = a0 + b0;      s0 -= (s0 >= 10) ? 10 : 0; // (a+b)%10
        int d0 = 10 + a0 - b0; d0 -= (d0 >= 10) ? 10 : 0; // (10+a-b)%10
        int s1 = a1 + b1;      s1 -= (s1 >= 10) ? 10 : 0;
        int d1 = 10 + a1 - b1; d1 -= (d1 >= 10) ? 10 : 0;

        f4 r;
        r.x = (float)(s0 + 1);
        r.y = (float)(d0 + 1);
        r.z = (float)(s1 + 1);
        r.w = (float)(d1 + 1);

        int g = (tile0 + it) * TPB + tid;
        if (g < nvec) {
            // write-once stream >> L2: non-temporal B128 store
            __builtin_nontemporal_store(r, (f4*)out + g);
        }
        buf ^= 1;
    }
}

__global__ __launch_bounds__(TPB)
void tail_copy(const float* __restrict__ x1, const float* __restrict__ x2,
               float* __restrict__ out, int B) {
    // B is a multiple of 4 (524288); stream x1 then x2 into the output tail.
    int i = (blockIdx.x * TPB + threadIdx.x) * 4;
    if (i < B) {
        f4 v = *(const f4*)(x1 + i);
        __builtin_nontemporal_store(v, (f4*)(out + i));
    } else if (i < 2 * B) {
        int j = i - B;
        f4 v = *(const f4*)(x2 + j);
        __builtin_nontemporal_store(v, (f4*)(out + i));
    }
}

extern "C" void kernel_launch(void* const* d_in, const int* in_sizes, int n_in,
                              void* d_out, int out_size, void* d_ws, size_t ws_size,
                              hipStream_t stream) {
    const int*   inp = (const int*)d_in[0];
    const float* x1  = (const float*)d_in[1];
    const float* x2  = (const float*)d_in[2];
    float*       out = (float*)d_out;

    const int nInts = in_sizes[0];        // B*S = 33,554,432
    const int B     = in_sizes[1];        // 524,288
    const int nvec  = nInts / 4;          // int4 groups

    const int tiles  = (nvec + TPB - 1) / TPB;
    const int blocks = (tiles + ITERS - 1) / ITERS;
    mixer_main<<<blocks, TPB, 0, stream>>>(inp, out, nvec);

    const int nTailVec = (2 * B + 3) / 4;
    const int tblocks  = (nTailVec + TPB - 1) / TPB;
    tail_copy<<<tblocks, TPB, 0, stream>>>(x1, x2, out + (size_t)nInts, B);
}